// RGCNEncoder_19628000543182
// MI455X (gfx1250) — compile-verified
//
#include <hip/hip_runtime.h>

// CDNA5 / gfx1250, wave32. fp32 WMMA GEMM (LDS-staged A, fragment-packed B)
// + atomic scatter RGCN with per-relation mean accumulators.

typedef float v2f __attribute__((ext_vector_type(2)));
typedef float v8f __attribute__((ext_vector_type(8)));

#define D 128
#define NREL 3
#define LDA_PAD 132           // LDS row stride (floats): 4m+kb+j covers 64 banks
#define PACKB_FLOATS (8 * 32 * 32 * 2)   // 16384 floats per packed 128x128 weight

// ---------------- degree count: deg[(dst+dstOff)*3 + et] += 1 ----------------
__global__ void k_deg(const int* __restrict__ dst, int dstOff, int et, int E,
                      int* __restrict__ deg) {
    int e = blockIdx.x * blockDim.x + threadIdx.x;
    if (e < E) atomicAdd(&deg[(dst[e] + dstOff) * NREL + et], 1);
}

// ------------- scatter: acc_r[dst] += x[src] / deg  (32 lanes per edge) -------------
__global__ void k_scatter(const int* __restrict__ src, const int* __restrict__ dst,
                          int srcOff, int dstOff, int et, int E,
                          const float* __restrict__ x, const int* __restrict__ deg,
                          float* __restrict__ accR) {
    long long tid = (long long)blockIdx.x * blockDim.x + threadIdx.x;
    int e = (int)(tid >> 5);
    int l = (int)(tid & 31);
    if (e >= E) return;
    int s = src[e] + srcOff;
    int d = dst[e] + dstOff;
    int dg = deg[d * NREL + et];
    float norm = 1.0f / (float)(dg > 1 ? dg : 1);
    const float4 xv = reinterpret_cast<const float4*>(x + (long long)s * D)[l];
    float* out = accR + (long long)d * D + l * 4;
    unsafeAtomicAdd(out + 0, xv.x * norm);
    unsafeAtomicAdd(out + 1, xv.y * norm);
    unsafeAtomicAdd(out + 2, xv.z * norm);
    unsafeAtomicAdd(out + 3, xv.w * norm);
}

// ---- pack a 128x128 row-major weight into per-wave WMMA-B fragment order ----
// Bp[((t*32 + k0)*32 + l)] = float2{ B[4k0+kb][16t+n], B[4k0+kb+1][16t+n] }
// with n = l%16, kb = (l/16)*2.  Inner-loop load becomes a coalesced b64.
__global__ void k_packB(const float* __restrict__ B, float* __restrict__ Bp) {
    int tid = blockIdx.x * blockDim.x + threadIdx.x;   // 8*32*32 = 8192 threads
    if (tid >= 8 * 32 * 32) return;
    int l  = tid & 31;
    int k0 = (tid >> 5) & 31;
    int t  = tid >> 10;
    int n  = l & 15;
    int kb = (l >> 4) << 1;
    int k  = 4 * k0 + kb;
    float2 v;
    v.x = B[(size_t)k * D + t * 16 + n];
    v.y = B[(size_t)(k + 1) * D + t * 16 + n];
    *reinterpret_cast<float2*>(Bp + (size_t)tid * 2) = v;
}

// ------------- fused multi-source GEMM: C = act( sum_s A_s @ B_s + bias ) -------------
// A_s: M x 128 row-major (M % 16 == 0), Bp_s: packed by k_packB, C: M x 128.
// Block = 8 wave32s covering a 16-row M tile; wave w owns N cols [16w,16w+16).
// A tiles staged in LDS (conflict-free), B read as coalesced packed b64 fragments.
template <int NSRC>
__global__ void __launch_bounds__(256)
k_gemm128(const float* A0, const float* __restrict__ Bp0,
          const float* A1, const float* __restrict__ Bp1,
          const float* A2, const float* __restrict__ Bp2,
          const float* A3, const float* __restrict__ Bp3,
          const float* __restrict__ bias, float* C, int do_relu) {
    __shared__ float lA[NSRC * 16 * LDA_PAD];

    const int tid  = threadIdx.x;
    const int lane = tid & 31;
    const int wave = tid >> 5;
    const int row  = blockIdx.x * 16;
    const int col  = wave * 16;
    const int m    = lane & 15;
    const int kb   = (lane >> 4) << 1;   // 0 or 2

    // ---- stage A tiles (16x128 each) into LDS, coalesced float4 ----
#pragma unroll
    for (int s = 0; s < NSRC; ++s) {
        const float* A = (s == 0) ? A0 : (s == 1) ? A1 : (s == 2) ? A2 : A3;
#pragma unroll
        for (int it = 0; it < 2; ++it) {
            int idx = it * 1024 + tid * 4;          // element in 16x128 tile
            int r   = idx >> 7;
            int kk  = idx & 127;
            float4 v = *reinterpret_cast<const float4*>(
                A + (size_t)(row + r) * D + kk);
            *reinterpret_cast<float4*>(&lA[s * 16 * LDA_PAD + r * LDA_PAD + kk]) = v;
        }
    }
    __syncthreads();

    v8f c = {};
#pragma unroll
    for (int s = 0; s < NSRC; ++s) {
        const float* __restrict__ Bp =
            (s == 0) ? Bp0 : (s == 1) ? Bp1 : (s == 2) ? Bp2 : Bp3;
        const float2* __restrict__ bptr =
            reinterpret_cast<const float2*>(Bp) + ((size_t)wave * 32) * 32 + lane;
        const float* lAs = &lA[s * 16 * LDA_PAD + m * LDA_PAD + kb];
#pragma unroll
        for (int k0 = 0; k0 < 32; ++k0) {
            const float2 af = *reinterpret_cast<const float2*>(lAs + k0 * 4);
            const float2 bf = bptr[(size_t)k0 * 32];
            v2f a; a.x = af.x; a.y = af.y;
            v2f b; b.x = bf.x; b.y = bf.y;
            c = __builtin_amdgcn_wmma_f32_16x16x4_f32(
                    false, a, false, b, (short)0, c, false, false);
        }
    }

    // In-place safe: any aliasing source (A3 == C) was consumed during staging.
    const int n  = col + m;
    const float bv = bias[n];
#pragma unroll
    for (int v = 0; v < 8; ++v) {
        int mr = row + ((lane < 16) ? v : v + 8);
        float val = c[v] + bv;
        if (do_relu) val = fmaxf(val, 0.0f);
        C[(size_t)mr * D + n] = val;
    }
}

extern "C" void kernel_launch(void* const* d_in, const int* in_sizes, int n_in,
                              void* d_out, int out_size, void* d_ws, size_t ws_size,
                              hipStream_t stream) {
    const float* x_user    = (const float*)d_in[0];
    const float* x_item    = (const float*)d_in[1];
    const int*   ei_ui     = (const int*)d_in[2];
    const int*   ei_iu     = (const int*)d_in[3];
    const int*   ei_uu     = (const int*)d_in[4];
    const float* win_user  = (const float*)d_in[5];
    const float* bin_user  = (const float*)d_in[6];
    const float* win_item  = (const float*)d_in[7];
    const float* bin_item  = (const float*)d_in[8];
    const float* W0        = (const float*)d_in[9];
    const float* root0     = (const float*)d_in[10];
    const float* bias0     = (const float*)d_in[11];
    const float* W1        = (const float*)d_in[12];
    const float* root1     = (const float*)d_in[13];
    const float* bias1     = (const float*)d_in[14];
    const float* wout_user = (const float*)d_in[15];
    const float* bout_user = (const float*)d_in[16];
    const float* wout_item = (const float*)d_in[17];
    const float* bout_item = (const float*)d_in[18];

    const int Nu  = in_sizes[0] / D;   // 50000 (multiple of 16)
    const int Ni  = in_sizes[1] / D;   // 50000
    const int N   = Nu + Ni;           // 100000
    const int Eui = in_sizes[2] / 2;
    const int Eiu = in_sizes[3] / 2;
    const int Euu = in_sizes[4] / 2;

    // ---- workspace layout ----
    char*  ws  = (char*)d_ws;
    float* h   = (float*)ws;                                  // N*D floats
    float* acc = (float*)(ws + (size_t)N * D * 4);            // 3*N*D floats
    float* bp  = (float*)(ws + (size_t)N * D * 4 * (1 + NREL));       // 12 packed mats
    int*   deg = (int*)(ws + (size_t)N * D * 4 * (1 + NREL)
                            + (size_t)12 * PACKB_FLOATS * 4);         // 3*N ints

    // packed weight slots
    float* p_win_u  = bp + 0  * (size_t)PACKB_FLOATS;
    float* p_win_i  = bp + 1  * (size_t)PACKB_FLOATS;
    float* p_W0[3]  = { bp + 2 * (size_t)PACKB_FLOATS, bp + 3 * (size_t)PACKB_FLOATS,
                        bp + 4 * (size_t)PACKB_FLOATS };
    float* p_root0  = bp + 5  * (size_t)PACKB_FLOATS;
    float* p_W1[3]  = { bp + 6 * (size_t)PACKB_FLOATS, bp + 7 * (size_t)PACKB_FLOATS,
                        bp + 8 * (size_t)PACKB_FLOATS };
    float* p_root1  = bp + 9  * (size_t)PACKB_FLOATS;
    float* p_wout_u = bp + 10 * (size_t)PACKB_FLOATS;
    float* p_wout_i = bp + 11 * (size_t)PACKB_FLOATS;

    // ---- pack all weights into WMMA fragment order ----
    k_packB<<<32, 256, 0, stream>>>(win_user, p_win_u);
    k_packB<<<32, 256, 0, stream>>>(win_item, p_win_i);
    for (int r = 0; r < NREL; ++r) {
        k_packB<<<32, 256, 0, stream>>>(W0 + (size_t)r * D * D, p_W0[r]);
        k_packB<<<32, 256, 0, stream>>>(W1 + (size_t)r * D * D, p_W1[r]);
    }
    k_packB<<<32, 256, 0, stream>>>(root0, p_root0);
    k_packB<<<32, 256, 0, stream>>>(root1, p_root1);
    k_packB<<<32, 256, 0, stream>>>(wout_user, p_wout_u);
    k_packB<<<32, 256, 0, stream>>>(wout_item, p_wout_i);

    const float* nf = nullptr;

    // ---- input MLPs: h = relu(x @ win + bin) ----
    k_gemm128<1><<<Nu / 16, 256, 0, stream>>>(x_user, p_win_u, nf, nf, nf, nf, nf, nf,
                                              bin_user, h, 1);
    k_gemm128<1><<<Ni / 16, 256, 0, stream>>>(x_item, p_win_i, nf, nf, nf, nf, nf, nf,
                                              bin_item, h + (size_t)Nu * D, 1);

    for (int layer = 0; layer < 2; ++layer) {
        const float* bias = layer ? bias1 : bias0;
        float* const* pW  = layer ? p_W1 : p_W0;
        float*        pR  = layer ? p_root1 : p_root0;

        hipMemsetAsync(acc, 0, (size_t)NREL * N * D * sizeof(float), stream);
        hipMemsetAsync(deg, 0, (size_t)NREL * N * sizeof(int), stream);

        // degree per (dst, etype)
        k_deg<<<(Eui + 255) / 256, 256, 0, stream>>>(ei_ui + Eui, Nu, 0, Eui, deg);
        k_deg<<<(Eiu + 255) / 256, 256, 0, stream>>>(ei_iu + Eiu, 0, 1, Eiu, deg);
        k_deg<<<(Euu + 255) / 256, 256, 0, stream>>>(ei_uu + Euu, 0, 2, Euu, deg);

        // per-relation mean-accumulate of raw features
        k_scatter<<<(unsigned)(((long long)Eui * 32 + 255) / 256), 256, 0, stream>>>(
            ei_ui, ei_ui + Eui, 0, Nu, 0, Eui, h, deg, acc);
        k_scatter<<<(unsigned)(((long long)Eiu * 32 + 255) / 256), 256, 0, stream>>>(
            ei_iu, ei_iu + Eiu, Nu, 0, 1, Eiu, h, deg, acc + (size_t)N * D);
        k_scatter<<<(unsigned)(((long long)Euu * 32 + 255) / 256), 256, 0, stream>>>(
            ei_uu, ei_uu + Euu, 0, 0, 2, Euu, h, deg, acc + 2 * (size_t)N * D);

        // fused: h = relu(acc0@W_r0 + acc1@W_r1 + acc2@W_r2 + h@root + bias)
        k_gemm128<4><<<N / 16, 256, 0, stream>>>(acc,                   pW[0],
                                                 acc + (size_t)N * D,   pW[1],
                                                 acc + 2*(size_t)N * D, pW[2],
                                                 h,                     pR,
                                                 bias, h, 1);
    }

    // ---- output heads (no relu) ----
    float* out = (float*)d_out;
    k_gemm128<1><<<Nu / 16, 256, 0, stream>>>(h, p_wout_u, nf, nf, nf, nf, nf, nf,
                                              bout_user, out, 0);
    k_gemm128<1><<<Ni / 16, 256, 0, stream>>>(h + (size_t)Nu * D, p_wout_i,
                                              nf, nf, nf, nf, nf, nf,
                                              bout_item, out + (size_t)Nu * D, 0);
}